// NeuralODETrajectory_36086315221618
// MI455X (gfx1250) — compile-verified
//
#include <hip/hip_runtime.h>
#include <cstddef>
#include <cstdint>

#define BATCH 1024
#define DIM   1024
#define NUM_TIMES 32
#define SUBSTEPS 2

typedef __attribute__((ext_vector_type(2))) float v2f;
typedef __attribute__((ext_vector_type(8))) float v8f;
typedef __attribute__((ext_vector_type(4))) unsigned int v4u;
typedef __attribute__((ext_vector_type(8))) int v8i;
typedef __attribute__((ext_vector_type(4))) int v4i;

#define KC   16   // K-chunk staged through LDS by the TDM
#define LDSP 20   // LDS row stride in floats = KC + 4 (TDM pad: 4 DWORDs / 16 DWORDs)
#define NCHUNK (DIM / KC)

// ---------------------------------------------------------------------------
// TDM: issue an async DMA of a [128 x KC] f32 tile (row stride DIM elements)
// from global memory into LDS at byte offset lds_off, with hardware padding of
// 4 DWORDs after every 16 DWORDs -> padded LDS row stride of LDSP floats.
// D# layout per CDNA5 ISA ch.8 (group0: count/lds/global/type, group1: dims).
// ---------------------------------------------------------------------------
__device__ __forceinline__ void tdm_load_tile(const float* gptr,
                                              unsigned lds_off) {
  const unsigned long long ga = (unsigned long long)(uintptr_t)gptr;
  v4u g0;
  g0[0] = 1u;                                        // count=1, flags=0
  g0[1] = lds_off;                                   // lds_addr (bytes)
  g0[2] = (unsigned)ga;                              // global_addr[31:0]
  g0[3] = (unsigned)((ga >> 32) & 0x01ffffffull)     // global_addr[56:32]
          | (2u << 30);                              // type=2 ("image")
  v8i g1;
  g1[0] = (int)((2u << 16)     // data_size = 4 bytes
                | (1u << 20)   // pad_enable
                | (3u << 22)   // pad_interval: 2^(3+1) = 16 DWORDs
                | (3u << 25)); // pad_amount:  3+1 = 4 DWORDs
  g1[1] = (int)((DIM & 0xffffu) << 16);              // tensor_dim0[15:0]
  g1[2] = (int)((DIM >> 16) | ((DIM & 0xffffu) << 16)); // dim0 hi | dim1 lo
  g1[3] = (int)((DIM >> 16) | ((unsigned)KC << 16)); // dim1 hi | tile_dim0=KC
  g1[4] = 128;                                       // tile_dim1=128, tile_dim2=0
  g1[5] = DIM;                                       // tensor_dim0_stride lo
  g1[6] = 0;                                         // stride0 hi | stride1 lo
  g1[7] = 0;
  v4i g2;
  g2[0] = 1; g2[1] = 1; g2[2] = 0; g2[3] = 0;        // tensor_dim2/3 = 1 (unused)
  v4i g3;
  g3[0] = 0; g3[1] = (int)(1u << 16); g3[2] = 0; g3[3] = 0; // tensor_dim4 = 1
#if __clang_major__ >= 23
  v8i gx = {0, 0, 0, 0, 0, 0, 0, 0};
  __builtin_amdgcn_tensor_load_to_lds(g0, g1, g2, g3, gx, 0);
#else
  __builtin_amdgcn_tensor_load_to_lds(g0, g1, g2, g3, 0);
#endif
}

// ---------------------------------------------------------------------------
// C[m,n] = sum_k A[m,k] * Wm[n,k] + bias[n]   (i.e. C = A * Wm^T + bias)
// A, Wm, C: 1024x1024 row-major f32.
// WG: 256 threads (8 waves), tile 128x128; wave tile 32x64 (2x4 WMMA tiles).
// TDM double-buffered K staging, V_WMMA_F32_16X16X4_F32 compute.
// ---------------------------------------------------------------------------
__global__ __launch_bounds__(256) void node_gemm_nt_bias(
    const float* __restrict__ A,
    const float* __restrict__ Wm,
    const float* __restrict__ bias,
    float* __restrict__ C)
{
  __shared__ float lA[2][128 * LDSP];
  __shared__ float lB[2][128 * LDSP];

  const int tid   = threadIdx.x;
  const int lane  = tid & 31;
  const int wave  = tid >> 5;
  const int wm    = wave & 3;    // wave M index (4 waves)
  const int wn    = wave >> 2;   // wave N index (2 waves)
  const int mWG   = blockIdx.y * 128;
  const int nWG   = blockIdx.x * 128;
  const int lmod  = lane & 15;
  const int lhalf = lane >> 4;

  const v8f vzero = {0.f, 0.f, 0.f, 0.f, 0.f, 0.f, 0.f, 0.f};
  v8f acc[2][4];
#pragma unroll
  for (int mt = 0; mt < 2; ++mt)
#pragma unroll
    for (int nt = 0; nt < 4; ++nt)
      acc[mt][nt] = vzero;

  // Prologue: wave 0 programs the TDM for the first K-chunk of both tiles.
  if (wave == 0) {
    tdm_load_tile(A  + (size_t)mWG * DIM, (unsigned)(uintptr_t)&lA[0][0]);
    tdm_load_tile(Wm + (size_t)nWG * DIM, (unsigned)(uintptr_t)&lB[0][0]);
  }

  for (int k = 0; k < NCHUNK; ++k) {
    const int cur = k & 1;
    __builtin_amdgcn_s_wait_tensorcnt(0);  // only wave 0 has outstanding TDM
    __syncthreads();                       // tile[cur] visible to all waves

    // Overlap: kick off the next chunk's DMA into the other buffer while the
    // whole WG runs WMMA on buf[cur]. Barrier above guarantees buf[1-cur] is
    // no longer being read.
    if (wave == 0 && (k + 1) < NCHUNK) {
      const int k0n = (k + 1) * KC;
      tdm_load_tile(A  + (size_t)mWG * DIM + k0n,
                    (unsigned)(uintptr_t)&lA[1 - cur][0]);
      tdm_load_tile(Wm + (size_t)nWG * DIM + k0n,
                    (unsigned)(uintptr_t)&lB[1 - cur][0]);
    }

    const float* bufA = &lA[cur][0];
    const float* bufB = &lB[cur][0];
#pragma unroll
    for (int ks = 0; ks < KC / 4; ++ks) {
      const int kk = ks * 4 + 2 * lhalf;   // K pair for this lane-half
      v2f afr[2];
#pragma unroll
      for (int mt = 0; mt < 2; ++mt)
        afr[mt] = *(const v2f*)(&bufA[(wm * 32 + mt * 16 + lmod) * LDSP + kk]);
#pragma unroll
      for (int nt = 0; nt < 4; ++nt) {
        const v2f bfr = *(const v2f*)(&bufB[(wn * 64 + nt * 16 + lmod) * LDSP + kk]);
#pragma unroll
        for (int mt = 0; mt < 2; ++mt)
          acc[mt][nt] = __builtin_amdgcn_wmma_f32_16x16x4_f32(
              false, afr[mt], false, bfr, (short)0, acc[mt][nt], false, false);
      }
    }
  }

  // Epilogue: C/D layout lane=(N, M-half), vgpr = M%8. Add bias[n].
#pragma unroll
  for (int nt = 0; nt < 4; ++nt) {
    const int col = nWG + wn * 64 + nt * 16 + lmod;
    const float bv = bias[col];
#pragma unroll
    for (int mt = 0; mt < 2; ++mt) {
      const int rbase = mWG + wm * 32 + mt * 16 + lhalf * 8;
#pragma unroll
      for (int v = 0; v < 8; ++v)
        C[(size_t)(rbase + v) * DIM + col] = acc[mt][nt][v] + bv;
    }
  }
}

// ---------------------------------------------------------------------------
// out = y + h * (c1*k1 + ... + c6*k6), h = (t[iv+1]-t[iv]) / SUBSTEPS.
// Coefficients are launch-uniform; zero-coeff terms are skipped (no read).
// ---------------------------------------------------------------------------
__global__ __launch_bounds__(256) void node_rk_combine(
    float* __restrict__ out, const float* __restrict__ y,
    const float* __restrict__ k1, const float* __restrict__ k2,
    const float* __restrict__ k3, const float* __restrict__ k4,
    const float* __restrict__ k5, const float* __restrict__ k6,
    float c1, float c2, float c3, float c4, float c5, float c6,
    const float* __restrict__ t_eval, int iv)
{
  const float h = (t_eval[iv + 1] - t_eval[iv]) * (1.0f / SUBSTEPS);
  const size_t i = (size_t)blockIdx.x * blockDim.x + threadIdx.x;
  float s = 0.0f;
  if (c1 != 0.0f) s += c1 * k1[i];
  if (c2 != 0.0f) s += c2 * k2[i];
  if (c3 != 0.0f) s += c3 * k3[i];
  if (c4 != 0.0f) s += c4 * k4[i];
  if (c5 != 0.0f) s += c5 * k5[i];
  if (c6 != 0.0f) s += c6 * k6[i];
  out[i] = y[i] + h * s;
}

__global__ __launch_bounds__(256) void node_copy(
    float* __restrict__ dst, const float* __restrict__ src)
{
  const size_t i = (size_t)blockIdx.x * blockDim.x + threadIdx.x;
  dst[i] = src[i];
}

// ---------------------------------------------------------------------------
static inline void launch_gemm(const float* A, const float* W, const float* b,
                               float* C, hipStream_t s) {
  dim3 grid(DIM / 128, BATCH / 128);
  node_gemm_nt_bias<<<grid, 256, 0, s>>>(A, W, b, C);
}

static inline void launch_comb(float* out, const float* y, float* const* k,
                               float c1, float c2, float c3, float c4, float c5,
                               float c6, const float* t_eval, int iv,
                               hipStream_t s) {
  node_rk_combine<<<(BATCH * DIM) / 256, 256, 0, s>>>(
      out, y, k[0], k[1], k[2], k[3], k[4], k[5], c1, c2, c3, c4, c5, c6,
      t_eval, iv);
}

extern "C" void kernel_launch(void* const* d_in, const int* in_sizes, int n_in,
                              void* d_out, int out_size, void* d_ws,
                              size_t ws_size, hipStream_t stream) {
  (void)in_sizes; (void)n_in; (void)out_size; (void)ws_size;
  const float* start  = (const float*)d_in[0];  // [1024,1024]
  const float* t_eval = (const float*)d_in[1];  // [32]
  const float* W      = (const float*)d_in[2];  // [1024,1024]
  const float* bias   = (const float*)d_in[3];  // [1024]
  float* out = (float*)d_out;                   // [32,1024,1024]
  float* ws  = (float*)d_ws;

  const size_t BD = (size_t)BATCH * DIM;
  float* k[6];
  for (int j = 0; j < 6; ++j) k[j] = ws + j * BD;
  float* ytmp   = ws + 6 * BD;  // state after substep 0
  float* ystage = ws + 7 * BD;  // RK stage input scratch

  // DOPRI5 tableau (5th-order solution).
  const float a21 = (float)(1.0 / 5.0);
  const float a31 = (float)(3.0 / 40.0),      a32 = (float)(9.0 / 40.0);
  const float a41 = (float)(44.0 / 45.0),     a42 = (float)(-56.0 / 15.0),
              a43 = (float)(32.0 / 9.0);
  const float a51 = (float)(19372.0 / 6561.0), a52 = (float)(-25360.0 / 2187.0),
              a53 = (float)(64448.0 / 6561.0), a54 = (float)(-212.0 / 729.0);
  const float a61 = (float)(9017.0 / 3168.0),  a62 = (float)(-355.0 / 33.0),
              a63 = (float)(46732.0 / 5247.0), a64 = (float)(49.0 / 176.0),
              a65 = (float)(-5103.0 / 18656.0);
  const float b1 = (float)(35.0 / 384.0),  b3 = (float)(500.0 / 1113.0),
              b4 = (float)(125.0 / 192.0), b5 = (float)(-2187.0 / 6784.0),
              b6 = (float)(11.0 / 84.0);

  // y(t0) = start_embedding
  node_copy<<<BD / 256, 256, 0, stream>>>(out, start);

  for (int iv = 0; iv < NUM_TIMES - 1; ++iv) {
    const float* ybase = out + (size_t)iv * BD;
    for (int ss = 0; ss < SUBSTEPS; ++ss) {
      launch_gemm(ybase, W, bias, k[0], stream);                       // k1
      launch_comb(ystage, ybase, k, a21, 0, 0, 0, 0, 0, t_eval, iv, stream);
      launch_gemm(ystage, W, bias, k[1], stream);                      // k2
      launch_comb(ystage, ybase, k, a31, a32, 0, 0, 0, 0, t_eval, iv, stream);
      launch_gemm(ystage, W, bias, k[2], stream);                      // k3
      launch_comb(ystage, ybase, k, a41, a42, a43, 0, 0, 0, t_eval, iv, stream);
      launch_gemm(ystage, W, bias, k[3], stream);                      // k4
      launch_comb(ystage, ybase, k, a51, a52, a53, a54, 0, 0, t_eval, iv, stream);
      launch_gemm(ystage, W, bias, k[4], stream);                      // k5
      launch_comb(ystage, ybase, k, a61, a62, a63, a64, a65, 0, t_eval, iv, stream);
      launch_gemm(ystage, W, bias, k[5], stream);                      // k6
      float* dst = (ss == SUBSTEPS - 1) ? (out + (size_t)(iv + 1) * BD) : ytmp;
      launch_comb(dst, ybase, k, b1, 0, b3, b4, b5, b6, t_eval, iv, stream);
      ybase = dst;
    }
  }
}